// GAT_72060961292603
// MI455X (gfx1250) — compile-verified
//
#include <hip/hip_runtime.h>
#include <math.h>

#define NNODES 100000
#define NEDGES 1600000
#define DIN    128
#define DOUT   32
#define NETOT  (NEDGES + NNODES)   // edges + self loops
#define LDA    136                 // padded row stride for A/U tiles (128 + 8 dwords)
#define WPB    4                   // waves per GEMM block

typedef __attribute__((ext_vector_type(2))) float    v2f;
typedef __attribute__((ext_vector_type(8))) float    v8f;
typedef __attribute__((ext_vector_type(4))) unsigned v4u;
typedef __attribute__((ext_vector_type(4))) int      v4i;
typedef __attribute__((ext_vector_type(8))) int      v8i;

// ---------------------------------------------------------------------------
// TDM: DMA a [rows x 128] f32 tile (row-major, stride 128) from global memory
// into LDS at lds_byte_off, padding each 128-DWORD row with 8 DWORDs so rows
// land at LDA=136 float stride. D# packing per CDNA5 ISA ch.8.
// ---------------------------------------------------------------------------
__device__ __forceinline__ void tdm_load_tile(unsigned lds_byte_off,
                                              const void* gptr,
                                              unsigned rows_avail)
{
    const unsigned long long ga = (unsigned long long)(uintptr_t)gptr;
    v4u g0;
    g0[0] = 1u;                                    // count=1, user descriptor
    g0[1] = lds_byte_off;                          // lds_addr [63:32]
    g0[2] = (unsigned)(ga & 0xFFFFFFFFu);          // global_addr [95:64]
    g0[3] = (unsigned)((ga >> 32) & 0x01FFFFFFu)   // global_addr [120:96]
          | 0x80000000u;                           // type=2 (image) [127:126]
    v8i g1;
    g1[0] = (int)((2u << 16)        // data_size = 4 bytes
                | (1u << 20)        // pad_enable
                | (6u << 22)        // pad_interval: every 128 DWORDs
                | (7u << 25));      // pad_amount: 8 DWORDs
    g1[1] = (int)(128u << 16);      // tensor_dim0 = 128   (bits [79:48])
    g1[2] = (int)(rows_avail << 16);// tensor_dim1         (bits [111:80])
    g1[3] = (int)(128u << 16);      // tile_dim0 = 128     (bits [127:112])
    g1[4] = WPB * 16;               // tile_dim1 = 64 rows; tile_dim2 = 0
    g1[5] = 128;                    // tensor_dim0_stride = 128 (bits [207:160])
    g1[6] = 0;
    g1[7] = 0;
    v4i z4 = {0, 0, 0, 0};
#if __clang_major__ >= 23
    v8i z8 = {};
    __builtin_amdgcn_tensor_load_to_lds(g0, g1, z4, z4, z8, 0);
#else
    __builtin_amdgcn_tensor_load_to_lds(g0, g1, z4, z4, 0);
#endif
}

// ---------------------------------------------------------------------------
// Kernel 1: fused dropout + dual GEMM (h_src = xd @ W_src, h_dst = xd @ W_dst)
// 128 threads = 4 waves; wave 0 TDM-loads the block's x / drop_u tiles into
// LDS, each wave then runs 16x32(x2) output tiles with V_WMMA_F32_16X16X4_F32
// (K=128 -> 32 steps, dropout fused at A-fragment read).
// ---------------------------------------------------------------------------
__global__ void __launch_bounds__(128) gat_gemm_kernel(
    const float* __restrict__ x, const float* __restrict__ drop_u,
    const float* __restrict__ Wsrc, const float* __restrict__ Wdst,
    float* __restrict__ h_src, float* __restrict__ h_dst)
{
    extern __shared__ float smem[];          // W[2*128*32] | A[64][LDA] | U[64][LDA]
    const int tid  = threadIdx.x;
    const int lane = tid & 31;
    const int wave = tid >> 5;

    float* Wl   = smem;
    float* Ablk = smem + 2 * DIN * DOUT;
    float* Ublk = Ablk + WPB * 16 * LDA;

    const int blockRow = blockIdx.x * (WPB * 16);

    if (wave == 0) {
        unsigned rows = (unsigned)(NNODES - blockRow);
        if (rows > WPB * 16) rows = WPB * 16;
        tdm_load_tile((unsigned)(uintptr_t)(void*)Ablk,
                      x + (size_t)blockRow * DIN, rows);
        tdm_load_tile((unsigned)(uintptr_t)(void*)Ublk,
                      drop_u + (size_t)blockRow * DIN, rows);
    }

    // Cooperative load of both weight matrices into LDS while TDM runs.
    for (int i = tid * 4; i < 2 * DIN * DOUT; i += 128 * 4) {
        const float* p = (i < DIN * DOUT) ? (Wsrc + i) : (Wdst + (i - DIN * DOUT));
        *(float4*)(Wl + i) = *(const float4*)p;
    }

    __builtin_amdgcn_s_wait_tensorcnt(0);    // wave 0 waits on its DMA; others no-op
    __syncthreads();

    const int tile = blockIdx.x * WPB + wave;
    if (tile >= NNODES / 16) return;         // 100000 % 16 == 0, no ragged M tiles

    const int m    = lane & 15;              // A-frag row / B-frag col
    const int koff = (lane >> 4) * 2;        // lanes 16-31 carry K+2, K+3
    const float* Ws   = Wl;                  // [128][32] row-major
    const float* Wd   = Wl + DIN * DOUT;
    const float* Arow = Ablk + (wave * 16 + m) * LDA + koff;
    const float* Urow = Ublk + (wave * 16 + m) * LDA + koff;

    v8f accS0 = {}, accS1 = {}, accD0 = {}, accD1 = {};
    #pragma unroll 4
    for (int k0 = 0; k0 < DIN; k0 += 4) {
        v2f a = *(const v2f*)(Arow + k0);
        v2f u = *(const v2f*)(Urow + k0);
        a.x = (u.x > 0.6f) ? a.x * 2.5f : 0.0f;   // dropout p=0.6, scale 1/0.4
        a.y = (u.y > 0.6f) ? a.y * 2.5f : 0.0f;
        const int kb = k0 + koff;
        v2f bs0 = { Ws[kb * DOUT + m],      Ws[(kb + 1) * DOUT + m]      };
        v2f bs1 = { Ws[kb * DOUT + 16 + m], Ws[(kb + 1) * DOUT + 16 + m] };
        v2f bd0 = { Wd[kb * DOUT + m],      Wd[(kb + 1) * DOUT + m]      };
        v2f bd1 = { Wd[kb * DOUT + 16 + m], Wd[(kb + 1) * DOUT + 16 + m] };
        accS0 = __builtin_amdgcn_wmma_f32_16x16x4_f32(false, a, false, bs0, (short)0, accS0, false, false);
        accS1 = __builtin_amdgcn_wmma_f32_16x16x4_f32(false, a, false, bs1, (short)0, accS1, false, false);
        accD0 = __builtin_amdgcn_wmma_f32_16x16x4_f32(false, a, false, bd0, (short)0, accD0, false, false);
        accD1 = __builtin_amdgcn_wmma_f32_16x16x4_f32(false, a, false, bd1, (short)0, accD1, false, false);
    }

    // C/D layout: VGPR i -> row = i + 8*(lane/16), col = lane%16 (+16 for tile 1)
    const int row0 = tile * 16 + (lane >> 4) * 8;
    #pragma unroll
    for (int i = 0; i < 8; ++i) {
        const size_t r = (size_t)(row0 + i) * DOUT;
        h_src[r + m]      = accS0[i];
        h_src[r + 16 + m] = accS1[i];
        h_dst[r + m]      = accD0[i];
        h_dst[r + 16 + m] = accD1[i];
    }
}

// ---------------------------------------------------------------------------
// Kernel 2: per-node attention logits  a_src = h_src . att_src  (wave per node)
// ---------------------------------------------------------------------------
__global__ void gat_att_kernel(const float* __restrict__ h_src,
                               const float* __restrict__ h_dst,
                               const float* __restrict__ att_src,
                               const float* __restrict__ att_dst,
                               float* __restrict__ a_src, float* __restrict__ a_dst)
{
    const int w    = (blockIdx.x * blockDim.x + threadIdx.x) >> 5;
    const int lane = threadIdx.x & 31;
    if (w >= NNODES) return;
    float vs = h_src[(size_t)w * DOUT + lane] * att_src[lane];
    float vd = h_dst[(size_t)w * DOUT + lane] * att_dst[lane];
    #pragma unroll
    for (int off = 16; off; off >>= 1) {
        vs += __shfl_xor(vs, off, 32);
        vd += __shfl_xor(vd, off, 32);
    }
    if (lane == 0) { a_src[w] = vs; a_dst[w] = vd; }
}

// ---------------------------------------------------------------------------
// Kernel 3: init accumulators (out_agg = bias_gat, e_max = -inf, denom = 0)
// ---------------------------------------------------------------------------
__global__ void gat_init_kernel(float* __restrict__ out_agg,
                                float* __restrict__ e_max,
                                float* __restrict__ denom,
                                const float* __restrict__ bias_gat)
{
    const int i = blockIdx.x * blockDim.x + threadIdx.x;
    if (i < NNODES * DOUT) out_agg[i] = bias_gat[i & 31];
    if (i < NNODES) {
        e_max[i] = __int_as_float(0xFF800000);  // -inf
        denom[i] = 0.0f;
    }
}

// Sign-aware ordered-int float atomic max (no NaNs in this workload).
__device__ __forceinline__ void atomicMaxF(float* addr, float v) {
    if (v >= 0.0f) atomicMax((int*)addr, __float_as_int(v));
    else           atomicMin((unsigned int*)addr, __float_as_uint(v));
}

__device__ __forceinline__ void edge_sd(int j, const int* __restrict__ ei, int& s, int& d) {
    if (j < NEDGES) { s = ei[j]; d = ei[NEDGES + j]; }
    else            { s = d = j - NEDGES; }         // self loops
}

// ---------------------------------------------------------------------------
// Kernel 4: segment max of leaky-relu edge scores (thread per edge)
// ---------------------------------------------------------------------------
__global__ void gat_edge_max_kernel(const int* __restrict__ ei,
                                    const float* __restrict__ a_src,
                                    const float* __restrict__ a_dst,
                                    float* __restrict__ e_max)
{
    const int j = blockIdx.x * blockDim.x + threadIdx.x;
    if (j >= NETOT) return;
    int s, d; edge_sd(j, ei, s, d);
    float e = a_src[s] + a_dst[d];
    e = (e > 0.0f) ? e : 0.2f * e;
    atomicMaxF(e_max + d, e);
}

// ---------------------------------------------------------------------------
// Kernel 5: e_exp = exp(e - e_max[dst]); denom[dst] += e_exp (thread per edge)
// ---------------------------------------------------------------------------
__global__ void gat_edge_exp_kernel(const int* __restrict__ ei,
                                    const float* __restrict__ a_src,
                                    const float* __restrict__ a_dst,
                                    const float* __restrict__ e_max,
                                    float* __restrict__ e_exp,
                                    float* __restrict__ denom)
{
    const int j = blockIdx.x * blockDim.x + threadIdx.x;
    if (j >= NETOT) return;
    int s, d; edge_sd(j, ei, s, d);
    float e = a_src[s] + a_dst[d];
    e = (e > 0.0f) ? e : 0.2f * e;
    const float ex = expf(e - e_max[d]);
    e_exp[j] = ex;
    atomicAdd(denom + d, ex);
}

// ---------------------------------------------------------------------------
// Kernel 6: alpha + scatter-add aggregation (wave per edge, lane per channel)
// 128B contiguous gather + 128B contiguous atomic burst per edge, L2 resident.
// ---------------------------------------------------------------------------
__global__ void gat_aggregate_kernel(const int* __restrict__ ei,
                                     const float* __restrict__ h_src,
                                     const float* __restrict__ e_exp,
                                     const float* __restrict__ denom,
                                     float* __restrict__ out_agg,
                                     float* __restrict__ alpha_out)
{
    const int w    = (blockIdx.x * blockDim.x + threadIdx.x) >> 5;
    const int lane = threadIdx.x & 31;
    if (w >= NETOT) return;
    int s, d; edge_sd(w, ei, s, d);
    const float alpha = e_exp[w] / denom[d];
    if (lane == 0) alpha_out[w] = alpha;
    atomicAdd(out_agg + (size_t)d * DOUT + lane,
              h_src[(size_t)s * DOUT + lane] * alpha);
}

// ---------------------------------------------------------------------------
// Kernel 7: residual linear + relu + log_softmax (wave per node, W_lin in LDS)
// ---------------------------------------------------------------------------
__global__ void __launch_bounds__(256) gat_final_kernel(
    const float* __restrict__ out_agg, const float* __restrict__ W_lin,
    const float* __restrict__ b_lin, float* __restrict__ logp)
{
    __shared__ float w[DOUT * DOUT];
    for (int i = threadIdx.x; i < DOUT * DOUT; i += 256) w[i] = W_lin[i];
    __syncthreads();

    const int node = (blockIdx.x * blockDim.x + threadIdx.x) >> 5;
    const int lane = threadIdx.x & 31;
    if (node >= NNODES) return;

    const float v = out_agg[(size_t)node * DOUT + lane];  // includes bias_gat
    float acc = v + b_lin[lane];
    #pragma unroll
    for (int k = 0; k < DOUT; ++k)
        acc = fmaf(__shfl(v, k, 32), w[k * DOUT + lane], acc);

    float t = fmaxf(acc, 0.0f);
    float m = t;
    #pragma unroll
    for (int off = 16; off; off >>= 1) m = fmaxf(m, __shfl_xor(m, off, 32));
    float ex = expf(t - m);
    float ssum = ex;
    #pragma unroll
    for (int off = 16; off; off >>= 1) ssum += __shfl_xor(ssum, off, 32);
    logp[(size_t)node * DOUT + lane] = t - m - logf(ssum);
}

// ---------------------------------------------------------------------------
extern "C" void kernel_launch(void* const* d_in, const int* in_sizes, int n_in,
                              void* d_out, int out_size, void* d_ws, size_t ws_size,
                              hipStream_t stream)
{
    (void)in_sizes; (void)n_in; (void)out_size; (void)ws_size;
    const float* x        = (const float*)d_in[0];
    const int*   ei       = (const int*)  d_in[1];
    const float* drop_u   = (const float*)d_in[2];
    const float* Wsrc     = (const float*)d_in[3];
    const float* Wdst     = (const float*)d_in[4];
    const float* att_src  = (const float*)d_in[5];
    const float* att_dst  = (const float*)d_in[6];
    const float* bias_gat = (const float*)d_in[7];
    const float* W_lin    = (const float*)d_in[8];
    const float* b_lin    = (const float*)d_in[9];

    float* logp      = (float*)d_out;
    float* alpha_out = (float*)d_out + (size_t)NNODES * DOUT;

    // Workspace layout (floats)
    float* ws      = (float*)d_ws;
    float* h_src   = ws;                    // 3,200,000
    float* h_dst   = ws + 3200000;          // 3,200,000
    float* a_src   = ws + 6400000;          //   100,000
    float* a_dst   = ws + 6500000;          //   100,000
    float* e_max   = ws + 6600000;          //   100,000
    float* denom   = ws + 6700000;          //   100,000
    float* e_exp   = ws + 6800000;          // 1,700,000
    float* out_agg = ws + 8500000;          // 3,200,000

    const int shmem = (2 * DIN * DOUT + 2 * WPB * 16 * LDA) * sizeof(float); // 102,400 B
    const int ntiles = NNODES / 16;                                          // 6250

    gat_init_kernel<<<(NNODES * DOUT + 255) / 256, 256, 0, stream>>>(
        out_agg, e_max, denom, bias_gat);

    gat_gemm_kernel<<<(ntiles + WPB - 1) / WPB, 128, shmem, stream>>>(
        x, drop_u, Wsrc, Wdst, h_src, h_dst);

    gat_att_kernel<<<(NNODES * 32 + 255) / 256, 256, 0, stream>>>(
        h_src, h_dst, att_src, att_dst, a_src, a_dst);

    gat_edge_max_kernel<<<(NETOT + 255) / 256, 256, 0, stream>>>(
        ei, a_src, a_dst, e_max);

    gat_edge_exp_kernel<<<(NETOT + 255) / 256, 256, 0, stream>>>(
        ei, a_src, a_dst, e_max, e_exp, denom);

    gat_aggregate_kernel<<<((size_t)NETOT * 32 + 255) / 256, 256, 0, stream>>>(
        ei, h_src, e_exp, denom, out_agg, alpha_out);

    gat_final_kernel<<<(NNODES * 32 + 255) / 256, 256, 0, stream>>>(
        out_agg, W_lin, b_lin, logp);
}